// Encoder_88235808129468
// MI455X (gfx1250) — compile-verified
//
#include <hip/hip_runtime.h>

// Reference dims (fixed by the harness)
#define Bc    64
#define Nc    512
#define DIN   256
#define DOUT  32
#define Hc    160
#define Wc    160
#define HWc   (Hc * Wc)        // 25600

// LDS row stride for transposed weight: 260 = 256+4, and 260 % 64 == 4
// -> a wave32 b64 read touches all 64 banks exactly once (conflict-free).
#define WSTRIDE 260

typedef __attribute__((ext_vector_type(2))) float v2f;
typedef __attribute__((ext_vector_type(8))) float v8f;

// ---------------------------------------------------------------------------
// Kernel 1: zero the 209.7 MB output with 128-bit stores (HBM-write bound).
// ---------------------------------------------------------------------------
__global__ __launch_bounds__(256) void zero_out_kernel(float4* __restrict__ out, long n4) {
    long i      = (long)blockIdx.x * blockDim.x + threadIdx.x;
    long stride = (long)gridDim.x * blockDim.x;
    float4 z = make_float4(0.f, 0.f, 0.f, 0.f);
    for (; i < n4; i += stride) out[i] = z;
}

// ---------------------------------------------------------------------------
// Kernel 2: fused  Linear(+bias) -> ReLU -> seq-mask -> scatter-add.
// One wave32 per 16x16 output tile; K=256 swept by v_wmma_f32_16x16x4_f32.
// Weight (256x32 fp32, 32KB) is staged TRANSPOSED in LDS once per block so
// the inner loop reads B as one conflict-free ds_load_b64 per step.
// ---------------------------------------------------------------------------
__global__ __launch_bounds__(256) void gemm_scatter_kernel(
    const float* __restrict__ A,     // [B*N, DIN]
    const float* __restrict__ Wt,    // [DIN, DOUT] row-major
    const float* __restrict__ bias,  // [DOUT]
    const int*   __restrict__ ex,
    const int*   __restrict__ ey,
    const int*   __restrict__ en,
    float*       __restrict__ out)   // [B, DOUT, H, W], pre-zeroed
{
    __shared__ float wT[DOUT * WSTRIDE];   // wT[n*WSTRIDE + k] = W[k][n]  (~33 KB)

    // ---- cooperative transposed fill: coalesced b128 reads of W ----------
    {
        const float4* w4 = (const float4*)Wt;          // 2048 float4s
        // 256 threads x 8 float4s each
        #pragma unroll
        for (int i = 0; i < 8; ++i) {
            int idx4 = threadIdx.x + i * 256;          // [0, 2048)
            float4 v = w4[idx4];
            int flat = idx4 << 2;                      // float index in W
            int k = flat >> 5;                         // DIN row
            int n = flat & 31;                         // DOUT col (n..n+3)
            wT[(n + 0) * WSTRIDE + k] = v.x;
            wT[(n + 1) * WSTRIDE + k] = v.y;
            wT[(n + 2) * WSTRIDE + k] = v.z;
            wT[(n + 3) * WSTRIDE + k] = v.w;
        }
    }
    __syncthreads();

    const int lane  = threadIdx.x & 31;
    const int wave  = threadIdx.x >> 5;
    const int tile  = blockIdx.x * 8 + wave;      // 4096 tiles total
    const int tileN = tile & 1;                   // 2 N-tiles (DOUT=32)
    const int tileM = tile >> 1;                  // 2048 M-tiles
    const int m0    = tileM << 4;
    const int n0    = tileN << 4;
    const int b     = m0 >> 9;                    // Nc=512 rows/batch; tile never straddles batches

    const int col   = lane & 15;                  // N within tile / A row selector
    const int khalf = lane >> 4;                  // K sub-pair: 0 -> K{0,1}, 1 -> K{2,3}

    // A: lane reads row (m0+col), K = k0 + khalf*2 + {0,1}  (aligned float2)
    const float* arow = A + (size_t)(m0 + col) * DIN + (khalf << 1);
    // B (from LDS, transposed): lane reads wT[n0+col][k0 + khalf*2 + {0,1}]
    const float* brow = wT + (size_t)(n0 + col) * WSTRIDE + (khalf << 1);

    v8f acc = {};
    #pragma unroll
    for (int k0 = 0; k0 < DIN; k0 += 4) {
        v2f av = *(const v2f*)(arow + k0);        // global b64
        v2f bv = *(const v2f*)(brow + k0);        // ds b64, conflict-free
        acc = __builtin_amdgcn_wmma_f32_16x16x4_f32(
                  false, av, false, bv, (short)0, acc, false, false);
    }

    // Epilogue: bias + ReLU + sequence mask + scatter-add.
    // C/D layout: VGPR r holds M = m0 + r + khalf*8, N = n0 + col.
    const int   c      = n0 + col;
    const float bv     = bias[c];
    const int   ecount = en[b];
    const size_t obase = ((size_t)b * DOUT + c) * HWc;

    #pragma unroll
    for (int r = 0; r < 8; ++r) {
        const int m = m0 + r + (khalf << 3);      // global row in [B*N)
        const int n = m & (Nc - 1);               // entity index within batch
        float v = acc[r] + bv;
        v = v > 0.f ? v : 0.f;                    // ReLU
        if (n < ecount) {                         // sequence mask
            const int fl = ey[m] * Wc + ex[m];    // y*W + x
            atomicAdd(out + obase + fl, v);       // global_atomic_add_f32
        }
    }
}

// ---------------------------------------------------------------------------
// Launcher
// ---------------------------------------------------------------------------
extern "C" void kernel_launch(void* const* d_in, const int* in_sizes, int n_in,
                              void* d_out, int out_size, void* d_ws, size_t ws_size,
                              hipStream_t stream) {
    const float* emb  = (const float*)d_in[0];   // [B, N, DIN]
    const float* w    = (const float*)d_in[1];   // [DIN, DOUT]
    const float* bias = (const float*)d_in[2];   // [DOUT]
    const int*   ex   = (const int*)  d_in[3];   // [B, N]
    const int*   ey   = (const int*)  d_in[4];   // [B, N]
    const int*   en   = (const int*)  d_in[5];   // [B]
    float*       out  = (float*)d_out;           // [B, DOUT, H, W]

    // 1) zero the scatter target (out_size = 52,428,800 floats, /4 exact)
    long n4 = (long)out_size >> 2;
    zero_out_kernel<<<2048, 256, 0, stream>>>((float4*)out, n4);

    // 2) fused GEMM + ReLU + mask + scatter-add
    //    4096 tiles, 8 waves/block -> 512 blocks of 256 threads (8 wave32)
    gemm_scatter_kernel<<<512, 256, 0, stream>>>(emb, w, bias, ex, ey, en, out);
}